// GraphEncoder_63574105915455
// MI455X (gfx1250) — compile-verified
//
#include <hip/hip_runtime.h>
#include <hip/hip_bf16.h>

// ---------------------------------------------------------------------------
// GNN encoder (NNConv x3 + pooling + MLPs) for MI455X / gfx1250.
// Dense GEMMs use V_WMMA_F32_16X16X4_F32 (native fp32 WMMA, wave32),
// fully unrolled with compile-time K/LDA/LDW so loads fold to immediate
// offsets. K<=32 GEMMs stash the whole B tile in VGPRs and sweep 5 row
// tiles per wave (B reuse), large-K GEMMs stream B.
// ---------------------------------------------------------------------------

#define NN   50000
#define EE   200000
#define BB   2048
#define HH   32
#define FLD  248          // padded feats leading dim (245 -> 248, K%4==0)
#define MAXN_INV 0.02f    // 1/50
#define EPS  1e-5f

typedef __attribute__((ext_vector_type(2))) float v2f;
typedef __attribute__((ext_vector_type(8))) float v8f;

// ------------------------------ helpers ------------------------------------

__device__ inline void atomicMaxF(float* addr, float v) {
    // Monotonic-bits trick; slots are initialized to -INF.
    if (v >= 0.0f) atomicMax((int*)addr, __float_as_int(v));
    else           atomicMin((unsigned int*)addr, (unsigned int)__float_as_int(v));
}

__global__ void k_zero(float* __restrict__ p, int n) {
    int i = blockIdx.x * blockDim.x + threadIdx.x;
    if (i < n) p[i] = 0.0f;
}

// feats layout per graph (FLD=248 cols, last 3 are zero padding):
//  [0] cnt/50 | [1..4] ec | [5..20] x_sum/50 | [21..36] x_mean | [37..52] x_max
//  then per layer l: [53+64l .. +31] mean(new_x), [53+64l+32 .. +63] max(new_x)
__global__ void k_init_feats(float* __restrict__ feats, int total) {
    int i = blockIdx.x * blockDim.x + threadIdx.x;
    if (i >= total) return;
    int c = i % FLD;
    bool is_max = (c >= 37 && c < 53) || (c >= 53 && c < 245 && ((c - 53) & 63) >= 32);
    feats[i] = is_max ? -__builtin_inff() : 0.0f;
}

// zero-pad fin0_w [245,256] into [248,256]
__global__ void k_pad_w(const float* __restrict__ w, float* __restrict__ wp, int total) {
    int i = blockIdx.x * blockDim.x + threadIdx.x;
    if (i >= total) return;
    int row = i >> 8;              // /256
    wp[i] = (row < 245) ? w[i] : 0.0f;
}

// ------------------------- graph statistics --------------------------------

__global__ void k_deg(const int* __restrict__ tgt, float* __restrict__ deg, int ne) {
    int e = blockIdx.x * blockDim.x + threadIdx.x;
    if (e < ne) atomicAdd(deg + tgt[e], 1.0f);
}

__global__ void k_pool_x(const float* __restrict__ x, const int* __restrict__ batch,
                         float* __restrict__ feats, float* __restrict__ cnt, int nn) {
    int n = blockIdx.x * blockDim.x + threadIdx.x;
    if (n >= nn) return;
    int b = batch[n];
    atomicAdd(cnt + b, 1.0f);
    float* f = feats + (size_t)b * FLD;
    for (int c = 0; c < 16; ++c) {
        float v = x[(size_t)n * 16 + c];
        atomicAdd(f + 5 + c, v);       // raw sum; finalize derives mean + /50
        atomicMaxF(f + 37 + c, v);
    }
}

__global__ void k_pool_edges(const int* __restrict__ src, const float* __restrict__ et,
                             const int* __restrict__ batch, float* __restrict__ feats, int ne) {
    int e = blockIdx.x * blockDim.x + threadIdx.x;
    if (e >= ne) return;
    int b = batch[src[e]];
    float* f = feats + (size_t)b * FLD;
    for (int t = 0; t < 4; ++t)
        atomicAdd(f + 1 + t, et[(size_t)e * 4 + t] * MAXN_INV);
}

// ------------------------- WMMA fp32 GEMM ----------------------------------
// One wave computes MT consecutive 16x16 C tiles down the M dimension:
//   C = A[M,K] @ W[K,*] (+bias, relu)
// A-tile 16x4: lanes 0-15 rows with K pair {k0,k0+1}; lanes 16-31 rows {k0+2,k0+3}.
// B-tile 4x16 mirrored; C/D: VGPR v -> M = v + 8*(lane>=16), N = lane&15.
// K, LDA, LDW compile-time: fully unrolled, loads use immediate offsets.
// MT>1: B tile (K/4 x v2f) stashed in registers, reused across row tiles.
template <int K, int LDA, int LDW, int MT, bool BIAS, bool RELU>
__global__ void k_gemm_wmma(const float* __restrict__ A,
                            const float* __restrict__ W,
                            const float* __restrict__ bias,
                            float* __restrict__ C, int ldc) {
    const int lane = threadIdx.x & 31;
    const int half = lane >> 4;
    const int lr   = lane & 15;
    const int col0 = blockIdx.y * 16;

    const float bv = BIAS ? bias[col0 + lr] : 0.0f;
    const float* __restrict__ Wp = W + (long)(2 * half) * LDW + col0 + lr;
    const float* __restrict__ Ap = A + 2 * half;

    if constexpr (MT == 1) {
        const int row0 = blockIdx.x * 16;
        v8f acc;
#pragma unroll
        for (int v = 0; v < 8; ++v) acc[v] = bv;
        const float* __restrict__ Ar = Ap + (long)(row0 + lr) * LDA;
#pragma unroll
        for (int k0 = 0; k0 < K; k0 += 4) {
            v2f a = *(const v2f*)(Ar + k0);
            v2f b;
            b.x = Wp[(long)k0 * LDW];
            b.y = Wp[(long)(k0 + 1) * LDW];
            acc = __builtin_amdgcn_wmma_f32_16x16x4_f32(
                      false, a, false, b, (short)0, acc, false, false);
        }
        if (RELU) {
#pragma unroll
            for (int v = 0; v < 8; ++v) acc[v] = fmaxf(acc[v], 0.0f);
        }
#pragma unroll
        for (int v = 0; v < 8; ++v)
            C[(long)(row0 + v + 8 * half) * ldc + col0 + lr] = acc[v];
    } else {
        // stash the whole B tile in registers once
        v2f Breg[K / 4];
#pragma unroll
        for (int k0 = 0; k0 < K; k0 += 4) {
            Breg[k0 / 4].x = Wp[(long)k0 * LDW];
            Breg[k0 / 4].y = Wp[(long)(k0 + 1) * LDW];
        }
#pragma unroll
        for (int m = 0; m < MT; ++m) {
            const int row0 = (blockIdx.x * MT + m) * 16;
            v8f acc;
#pragma unroll
            for (int v = 0; v < 8; ++v) acc[v] = bv;
            const float* __restrict__ Ar = Ap + (long)(row0 + lr) * LDA;
#pragma unroll
            for (int k0 = 0; k0 < K; k0 += 4) {
                v2f a = *(const v2f*)(Ar + k0);
                acc = __builtin_amdgcn_wmma_f32_16x16x4_f32(
                          false, a, false, Breg[k0 / 4], (short)0, acc, false, false);
            }
            if (RELU) {
#pragma unroll
                for (int v = 0; v < 8; ++v) acc[v] = fmaxf(acc[v], 0.0f);
            }
#pragma unroll
            for (int v = 0; v < 8; ++v)
                C[(long)(row0 + v + 8 * half) * ldc + col0 + lr] = acc[v];
        }
    }
}

// ------------------------- per-edge message --------------------------------
// msg[e,k] = sum_t et[e,t]*q_t[src,k] + r[src,k];  agg[tgt,k] += msg
// q layout: [N,160] = [q0|q1|q2|q3|r]
__global__ void k_edge_msg(const int* __restrict__ src, const int* __restrict__ tgt,
                           const float* __restrict__ et, const float* __restrict__ q,
                           float* __restrict__ agg, int ne) {
    int t = blockIdx.x * blockDim.x + threadIdx.x;
    int e = t >> 5;
    if (e >= ne) return;
    int k = t & 31;
    int s = src[e], d = tgt[e];
    const float* qs = q + (size_t)s * 160;
    const float* w  = et + (size_t)e * 4;
    float m = w[0] * qs[k] + w[1] * qs[32 + k] + w[2] * qs[64 + k]
            + w[3] * qs[96 + k] + qs[128 + k];
    atomicAdd(agg + (size_t)d * 32 + k, m);
}

// ----------------- combine: scatter-mean, pool, residual, BN stats ---------
__global__ void k_combine(const float* __restrict__ newx, const float* __restrict__ agg,
                          const float* __restrict__ deg, const int* __restrict__ batch,
                          float* __restrict__ h, float* __restrict__ feats,
                          float* __restrict__ bnS, float* __restrict__ bnQ,
                          int l, int nn) {
    __shared__ float sS[32];
    __shared__ float sQ[32];
    int t = threadIdx.x;
    if (t < 32) { sS[t] = 0.0f; sQ[t] = 0.0f; }
    __syncthreads();

    int n = blockIdx.x * blockDim.x + t;
    if (n < nn) {
        int b = batch[n];
        float invd = 1.0f / fmaxf(deg[n], 1.0f);
        float* fb = feats + (size_t)b * FLD + 53 + 64 * l;
        for (int i = 0; i < 32; ++i) {
            int k = (i + t) & 31;                    // rotate to spread LDS atomics
            size_t idx = (size_t)n * 32 + k;
            float v = newx[idx] + agg[idx] * invd;   // new_x (pre-activation)
            atomicAdd(fb + k, v);                    // mean-pool accumulator
            atomicMaxF(fb + 32 + k, v);              // max-pool
            float hn = fmaxf(v, 0.0f) + h[idx];      // ReLU + residual
            h[idx] = hn;
            atomicAdd(&sS[k], hn);
            atomicAdd(&sQ[k], hn * hn);
        }
    }
    __syncthreads();
    if (t < 32) { atomicAdd(&bnS[t], sS[t]); atomicAdd(&bnQ[t], sQ[t]); }
}

__global__ void k_bn(float* __restrict__ h, const float* __restrict__ bnS,
                     const float* __restrict__ bnQ, const float* __restrict__ gamma,
                     const float* __restrict__ beta, int l, int total) {
    int i = blockIdx.x * blockDim.x + threadIdx.x;
    if (i >= total) return;
    int k = i & 31;
    float mu  = bnS[k] * (1.0f / (float)NN);
    float var = bnQ[k] * (1.0f / (float)NN) - mu * mu;
    float s   = rsqrtf(var + EPS);
    h[i] = (h[i] - mu) * s * gamma[l * 32 + k] + beta[l * 32 + k];
}

// ------------------------------- finalize ----------------------------------
__global__ void k_finalize(float* __restrict__ feats, const float* __restrict__ cnt, int nb) {
    int b = blockIdx.x * blockDim.x + threadIdx.x;
    if (b >= nb) return;
    float c = cnt[b];
    float invc = 1.0f / fmaxf(c, 1.0f);
    bool empty = (c < 0.5f);
    float* f = feats + (size_t)b * FLD;
    f[0] = c * MAXN_INV;
    for (int j = 0; j < 16; ++j) {
        float raw = f[5 + j];
        f[21 + j] = raw * invc;        // x_mean
        f[5 + j]  = raw * MAXN_INV;    // x_sum / 50
        if (empty) f[37 + j] = 0.0f;   // empty-segment max -> 0
    }
    for (int l = 0; l < 3; ++l) {
        float* g = f + 53 + 64 * l;
        for (int j = 0; j < 32; ++j) {
            g[j] *= invc;                       // mean(new_x)
            if (empty) g[32 + j] = 0.0f;        // empty-segment max -> 0
        }
    }
}

// ------------------------------- host --------------------------------------

extern "C" void kernel_launch(void* const* d_in, const int* in_sizes, int n_in,
                              void* d_out, int out_size, void* d_ws, size_t ws_size,
                              hipStream_t stream) {
    const float* x       = (const float*)d_in[0];
    const int*   eidx    = (const int*)d_in[1];
    const float* et      = (const float*)d_in[2];
    const int*   batch   = (const int*)d_in[3];
    const float* mlp0_w  = (const float*)d_in[4];
    const float* mlp0_b  = (const float*)d_in[5];
    const float* mlp1_w  = (const float*)d_in[6];
    const float* mlp1_b  = (const float*)d_in[7];
    const float* enw     = (const float*)d_in[8];   // [3,4,1024]
    const float* enb     = (const float*)d_in[9];   // [3,1024]
    const float* rootw   = (const float*)d_in[10];  // [3,32,32]
    const float* convb   = (const float*)d_in[11];  // [3,32]
    const float* bng     = (const float*)d_in[12];
    const float* bnb     = (const float*)d_in[13];
    const float* fin0_w  = (const float*)d_in[14];  // [245,256]
    const float* fin0_b  = (const float*)d_in[15];
    const float* fin1_w  = (const float*)d_in[16];  // [256,128]
    const float* fin1_b  = (const float*)d_in[17];
    float*       out     = (float*)d_out;

    const int* src = eidx;
    const int* tgt = eidx + EE;

    // -------- workspace carve-out (floats) --------
    float* ws = (float*)d_ws;
    size_t off = 0;
    float* feats  = ws + off; off += (size_t)BB * FLD;
    float* cnt    = ws + off; off += BB;
    float* deg    = ws + off; off += NN;
    float* h      = ws + off; off += (size_t)NN * HH;
    float* newx   = ws + off; off += (size_t)NN * HH;   // also hpre of init MLP
    float* agg    = ws + off; off += (size_t)NN * HH;
    float* q      = ws + off; off += (size_t)NN * 160;
    float* bn     = ws + off; off += 64;                // [sum|sumsq]
    float* hidden = ws + off; off += (size_t)BB * 256;
    float* wpad   = ws + off; off += (size_t)FLD * 256; // zero-padded fin0_w
    (void)ws_size; (void)in_sizes; (void)n_in; (void)out_size;

    const int TB = 256;
    auto nb = [](long n, int b) { return (int)((n + b - 1) / b); };

    // init
    k_init_feats<<<nb((long)BB * FLD, TB), TB, 0, stream>>>(feats, BB * FLD);
    k_zero<<<nb(BB, TB), TB, 0, stream>>>(cnt, BB);
    k_zero<<<nb(NN, TB), TB, 0, stream>>>(deg, NN);
    k_pad_w<<<nb((long)FLD * 256, TB), TB, 0, stream>>>(fin0_w, wpad, FLD * 256);

    // graph statistics
    k_deg<<<nb(EE, TB), TB, 0, stream>>>(tgt, deg, EE);
    k_pool_x<<<nb(NN, TB), TB, 0, stream>>>(x, batch, feats, cnt, NN);
    k_pool_edges<<<nb(EE, TB), TB, 0, stream>>>(src, et, batch, feats, EE);

    // node GEMM grid: 50000 rows = 625 blocks * 5 row-tiles * 16
    dim3 gN(NN / (16 * 5), 2);

    // initial MLP: hpre = relu(x@W0+b0); h = hpre@W1+b1
    k_gemm_wmma<16, 16, 32, 5, true, true><<<gN, 32, 0, stream>>>(x, mlp0_w, mlp0_b, newx, 32);
    k_gemm_wmma<32, 32, 32, 5, true, false><<<gN, 32, 0, stream>>>(newx, mlp1_w, mlp1_b, h, 32);

    for (int l = 0; l < 3; ++l) {
        // q_t = h @ Wnn_t (t<4); r = h @ Bmat  -> q[N,160]
        for (int t = 0; t < 5; ++t) {
            const float* W = (t < 4) ? (enw + (size_t)l * 4096 + (size_t)t * 1024)
                                     : (enb + (size_t)l * 1024);
            k_gemm_wmma<32, 32, 32, 5, false, false><<<gN, 32, 0, stream>>>(
                h, W, nullptr, q + t * 32, 160);
        }
        k_zero<<<nb((long)NN * 32, TB), TB, 0, stream>>>(agg, NN * 32);
        k_zero<<<1, 64, 0, stream>>>(bn, 64);
        k_edge_msg<<<nb((long)EE * 32, TB), TB, 0, stream>>>(src, tgt, et, q, agg, EE);
        // newx = h @ root_w[l] + conv_b[l]
        k_gemm_wmma<32, 32, 32, 5, true, false><<<gN, 32, 0, stream>>>(
            h, rootw + (size_t)l * 1024, convb + (size_t)l * 32, newx, 32);
        k_combine<<<nb(NN, 128), 128, 0, stream>>>(newx, agg, deg, batch, h, feats,
                                                   bn, bn + 32, l, NN);
        k_bn<<<nb((long)NN * 32, TB), TB, 0, stream>>>(h, bn, bn + 32, bng, bnb, l, NN * 32);
    }

    k_finalize<<<nb(BB, TB), TB, 0, stream>>>(feats, cnt, BB);

    // final MLP: hidden = relu(feats@fin0+b0); out = hidden@fin1+b1 (split halves)
    dim3 gF0(BB / 16, 16);
    k_gemm_wmma<FLD, FLD, 256, 1, true, true><<<gF0, 32, 0, stream>>>(
        feats, wpad, fin0_b, hidden, 256);
    dim3 gF1(BB / 16, 4);
    k_gemm_wmma<256, 256, 128, 1, true, false><<<gF1, 32, 0, stream>>>(
        hidden, fin1_w, fin1_b, out, 64);
    k_gemm_wmma<256, 256, 128, 1, true, false><<<gF1, 32, 0, stream>>>(
        hidden, fin1_w + 64, fin1_b + 64, out + (size_t)BB * 64, 64);
}